// NCCForces_84619445665875
// MI455X (gfx1250) — compile-verified
//
#include <hip/hip_runtime.h>

// NCC forces, fused single-pass kernel for MI455X (gfx1250, wave32).
// Box sums done separably: Z pass = VALU sliding window over an LDS ring of
// raw slices; Y and X passes = banded-matrix GEMMs via V_WMMA_F32_16X16X4_F32.

typedef __attribute__((ext_vector_type(2))) float v2f;
typedef __attribute__((ext_vector_type(8))) float v8f;

#define NI 192
#define TS 16          // output tile edge (x and y)
#define HS 20          // halo'd slice edge (TS + 2*radius)
#define SL (HS * HS)   // 400 elements per slice
#define ZC 24          // z-slices per block
#define NPIX 125.0f
#define EPS 1e-5f

__device__ __forceinline__ int gidx(int z, int y, int x) {
    return (z * NI + y) * NI + x;
}

__global__ __launch_bounds__(256) void ncc_forces_kernel(
    const float* __restrict__ M, const float* __restrict__ F,
    const unsigned char* __restrict__ MMask, const unsigned char* __restrict__ FMask,
    float* __restrict__ out)
{
    // LDS: ring of 5 raw halo'd slices (m,f), z-summed quantities, y-summed, final box.
    __shared__ float rm[5 * SL];            // raw moving slices (ring)
    __shared__ float rf[5 * SL];            // raw fixed  slices (ring)
    __shared__ float zs[5 * SL];            // 5 quantities after Z pass (20x20)
    __shared__ float ys[5 * TS * HS];       // after Y pass (16 rows x 20 cols)
    __shared__ float bx[5 * TS * TS];       // after X pass (16x16)

    const int tid  = threadIdx.x;
    const int lane = tid & 31;
    const int wv   = tid >> 5;              // wave id (wave32)
    const int half = lane >> 4;             // 0 = lanes 0-15, 1 = lanes 16-31
    const int m16  = lane & 15;
    const int x0t  = blockIdx.x * TS;
    const int y0t  = blockIdx.y * TS;
    const int z0   = blockIdx.z * ZC;

    auto load_slice = [&](int gz) {
        int slot = ((gz % 5) + 5) % 5;
        for (int i = tid; i < SL; i += 256) {
            int ly = i / HS, lx = i - ly * HS;
            int gy = y0t + ly - 2, gx = x0t + lx - 2;
            float mv = 0.f, fv = 0.f;   // zero padding == conv 'SAME'
            if ((unsigned)gz < NI && (unsigned)gy < NI && (unsigned)gx < NI) {
                int id = gidx(gz, gy, gx);
                mv = M[id]; fv = F[id];
            }
            rm[slot * SL + i] = mv;
            rf[slot * SL + i] = fv;
        }
    };

    load_slice(z0 - 2); load_slice(z0 - 1); load_slice(z0); load_slice(z0 + 1);

    for (int z = z0; z < z0 + ZC; ++z) {
        load_slice(z + 2);
        __syncthreads();

        const int s0 = ((z - 2) % 5 + 5) % 5;
        const int s1 = ((z - 1) % 5 + 5) % 5;
        const int s2 = z % 5;
        const int s3 = (z + 1) % 5;
        const int s4 = (z + 2) % 5;
        (void)s0;

        // ---- Z pass: 5-tap sums of m, f, m*m, f*f, m*f on the 20x20 slab ----
        {
            const int ss[5] = {s0, s1, s2, s3, s4};
            for (int i = tid; i < SL; i += 256) {
                float sm = 0.f, sf = 0.f, smm = 0.f, sff = 0.f, smf = 0.f;
                #pragma unroll
                for (int k = 0; k < 5; ++k) {
                    float mv = rm[ss[k] * SL + i];
                    float fv = rf[ss[k] * SL + i];
                    sm += mv; sf += fv;
                    smm += mv * mv; sff += fv * fv; smf += mv * fv;
                }
                zs[0 * SL + i] = sm;  zs[1 * SL + i] = sf;
                zs[2 * SL + i] = smm; zs[3 * SL + i] = sff; zs[4 * SL + i] = smf;
            }
        }
        __syncthreads();

        // ---- Y pass via WMMA: out(16y x 16x) = Band(16x20) * zs(20x16) ----
        // 10 wave-tasks: 5 quantities x 2 x-blocks (x0=0 covers x0..15, x0=4 covers x16..19).
        for (int task = wv; task < 10; task += 8) {
            const int q = task >> 1, xb = task & 1, x0 = xb * 4;
            v8f d = {};
            #pragma unroll
            for (int c = 0; c < 5; ++c) {
                const int K = 4 * c + 2 * half;    // A/B wave32 layout: K = 4c + j + 2*half
                v2f a, b;
                a.x = (K     >= m16 && K     <= m16 + 4) ? 1.f : 0.f;  // Band[y][k]
                a.y = (K + 1 >= m16 && K + 1 <= m16 + 4) ? 1.f : 0.f;
                b.x = zs[q * SL + K       * HS + x0 + m16];            // zs rows
                b.y = zs[q * SL + (K + 1) * HS + x0 + m16];
                d = __builtin_amdgcn_wmma_f32_16x16x4_f32(false, a, false, b,
                                                          (short)0, d, false, false);
            }
            #pragma unroll
            for (int j = 0; j < 8; ++j) {
                const int yo = j + 8 * half;       // D layout: VGPR j -> M=j / M=j+8
                const int xg = x0 + m16;
                if (xb == 0 || m16 >= 12)          // xb1 contributes x=16..19 only
                    ys[q * TS * HS + yo * HS + xg] = d[j];
            }
        }
        __syncthreads();

        // ---- X pass via WMMA: out(16y x 16x) = ys(16x20) * Band(20x16) ----
        if (wv < 5) {
            const int q = wv;
            v8f d = {};
            #pragma unroll
            for (int c = 0; c < 5; ++c) {
                const int K = 4 * c + 2 * half;
                v2f a, b;
                a.x = ys[q * TS * HS + m16 * HS + K];
                a.y = ys[q * TS * HS + m16 * HS + K + 1];
                b.x = (K     >= m16 && K     <= m16 + 4) ? 1.f : 0.f;  // Band[k][x]
                b.y = (K + 1 >= m16 && K + 1 <= m16 + 4) ? 1.f : 0.f;
                d = __builtin_amdgcn_wmma_f32_16x16x4_f32(false, a, false, b,
                                                          (short)0, d, false, false);
            }
            #pragma unroll
            for (int j = 0; j < 8; ++j)
                bx[q * 256 + (j + 8 * half) * 16 + m16] = d[j];
        }
        __syncthreads();

        // ---- Combine (one output voxel per thread) ----
        {
            const int y = tid >> 4, x = tid & 15;
            const int gy = y0t + y, gx = x0t + x;

            const float sum_m  = bx[tid];
            const float sum_f  = bx[256 + tid];
            const float sum_mm = bx[512 + tid];
            const float sum_ff = bx[768 + tid];
            const float sum_mf = bx[1024 + tid];

            const float inv = 1.0f / NPIX;
            const float mean_m = sum_m * inv, mean_f = sum_f * inv;
            const float var_m = sum_mm - 2.f * mean_m * sum_m + NPIX * mean_m * mean_m;
            const float var_f = sum_ff - 2.f * mean_f * sum_f + NPIX * mean_f * mean_f;
            const float var_mf = var_m * var_f;
            const float cross = sum_mf - mean_f * sum_m - mean_m * sum_f
                              + NPIX * mean_m * mean_f;

            const int cyx = (y + 2) * HS + (x + 2);
            const float mc = rm[s2 * SL + cyx] - mean_m;
            const float fc = rf[s2 * SL + cyx] - mean_f;

            float factor = 0.f;
            if ((var_mf > EPS) && (var_f > EPS) && (fc != 0.f) && (mc != 0.f))
                factor = 2.f * cross / var_mf * (mc - cross * fc / var_f);

            // gradients (central interior, one-sided at volume edges)
            float gmz, gfz, gmy, gfy, gmx, gfx;
            if (z == 0)            { gmz = rm[s3*SL+cyx] - rm[s2*SL+cyx];
                                     gfz = rf[s3*SL+cyx] - rf[s2*SL+cyx]; }
            else if (z == NI - 1)  { gmz = rm[s2*SL+cyx] - rm[s1*SL+cyx];
                                     gfz = rf[s2*SL+cyx] - rf[s1*SL+cyx]; }
            else                   { gmz = 0.5f * (rm[s3*SL+cyx] - rm[s1*SL+cyx]);
                                     gfz = 0.5f * (rf[s3*SL+cyx] - rf[s1*SL+cyx]); }
            const int b = s2 * SL;
            if (gy == 0)           { gmy = rm[b+cyx+HS] - rm[b+cyx];
                                     gfy = rf[b+cyx+HS] - rf[b+cyx]; }
            else if (gy == NI - 1) { gmy = rm[b+cyx] - rm[b+cyx-HS];
                                     gfy = rf[b+cyx] - rf[b+cyx-HS]; }
            else                   { gmy = 0.5f * (rm[b+cyx+HS] - rm[b+cyx-HS]);
                                     gfy = 0.5f * (rf[b+cyx+HS] - rf[b+cyx-HS]); }
            if (gx == 0)           { gmx = rm[b+cyx+1] - rm[b+cyx];
                                     gfx = rf[b+cyx+1] - rf[b+cyx]; }
            else if (gx == NI - 1) { gmx = rm[b+cyx] - rm[b+cyx-1];
                                     gfx = rf[b+cyx] - rf[b+cyx-1]; }
            else                   { gmx = 0.5f * (rm[b+cyx+1] - rm[b+cyx-1]);
                                     gfx = 0.5f * (rf[b+cyx+1] - rf[b+cyx-1]); }

            const int id = gidx(z, gy, gx);
            const float u = (MMask[id] | FMask[id]) ? 1.f : 0.f;
            const float s = -factor * 0.5f * u;
            const int V = NI * NI * NI;
            out[id]          = s * (gmz + gfz);
            out[V + id]      = s * (gmy + gfy);
            out[2 * V + id]  = s * (gmx + gfx);
        }
        // no trailing barrier needed: next iteration's load touches only the
        // retiring ring slot ((z+3)%5 == (z-2)%5), which combine never reads,
        // and the post-load barrier orders all zs/ys/bx writes.
    }
}

extern "C" void kernel_launch(void* const* d_in, const int* in_sizes, int n_in,
                              void* d_out, int out_size, void* d_ws, size_t ws_size,
                              hipStream_t stream) {
    const float* M = (const float*)d_in[0];
    const float* F = (const float*)d_in[1];
    const unsigned char* MM = (const unsigned char*)d_in[2];
    const unsigned char* FM = (const unsigned char*)d_in[3];
    // d_in[4] (original_image_spacing) is unused: reference assumes unit spacing.
    dim3 grid(NI / TS, NI / TS, NI / ZC);   // 12 x 12 x 8 blocks, 256 thr (8 wave32)
    ncc_forces_kernel<<<grid, 256, 0, stream>>>(M, F, MM, FM, (float*)d_out);
}